// MLP_RSNA5_73778948210888
// MI455X (gfx1250) — compile-verified
//
#include <hip/hip_runtime.h>

typedef __attribute__((ext_vector_type(2))) float v2f;
typedef __attribute__((ext_vector_type(8))) float v8f;

constexpr int Bb   = 16384;
constexpr int Gg   = 25;
constexpr int Dd   = 128;
constexpr int Hh   = 256;
constexpr int Oo   = 3;
constexpr int OUTC = 75;

constexpr int NG   = 5;        // groups per block
constexpr int ROWS = 128;      // batch rows per block (8 waves x 16)
constexpr int NT   = Hh / 16;  // 16 N-tiles of the hidden dim
constexpr int NK   = Dd / 4;   // 32 k-steps per WMMA chain

// Dynamic LDS layout (bytes)
constexpr size_t LDS_W1  = (size_t)NT * NK * 32 * 8;  // 131072: W1 in fragment order (float2)
constexpr size_t LDS_SK  = (size_t)NG * Dd * 4;       //   2560: gather indices for NG groups
constexpr size_t LDS_B1  = (size_t)Hh * 4;            //   1024: b1
constexpr size_t LDS_W2  = (size_t)Hh * Oo * 4;       //   3072: W2
constexpr size_t LDS_TOT = LDS_W1 + LDS_SK + LDS_B1 + LDS_W2;  // ~134.5 KB -> 2 blocks/WGP

__global__ void zero_out_kernel(float* __restrict__ out, int n) {
    int i = blockIdx.x * blockDim.x + threadIdx.x;
    if (i < n) out[i] = 0.0f;
}

__global__ __launch_bounds__(256)
void mlp_wmma_f32_kernel(const float* __restrict__ x,
                         const float* __restrict__ W1,
                         const float* __restrict__ b1,
                         const float* __restrict__ W2,
                         const float* __restrict__ b2,
                         const int*   __restrict__ k_idx,
                         const int*   __restrict__ v_idx,
                         float*       __restrict__ out) {
    extern __shared__ char smem[];
    v2f*   w1f = (v2f*)smem;                                   // [NT][NK][32] B-fragments
    int*   sk  = (int*)(smem + LDS_W1);                        // [NG][Dd]
    float* sb1 = (float*)(smem + LDS_W1 + LDS_SK);             // [Hh]
    float* sw2 = (float*)(smem + LDS_W1 + LDS_SK + LDS_B1);    // [Hh][Oo]

    const int tid   = threadIdx.x;
    const int gbase = blockIdx.y * NG;

    // ---- Stage W1 into LDS in exact per-lane WMMA B-fragment order ----
    // Fragment (t, kk, lane):  kb = 4*kk + (lane<16 ? 0 : 2), n = 16*t + (lane&15)
    //   value = { W1[kb][n], W1[kb+1][n] }
    // Entry e = (t*NK + kk)*32 + lane; each W1 element appears exactly once.
#pragma unroll 4
    for (int i = 0; i < (NT * NK * 32) / 256; ++i) {
        const int e   = tid + i * 256;
        const int t   = e >> 10;
        const int kk  = (e >> 5) & 31;
        const int ln  = e & 31;
        const int c0  = (ln & 16) ? 2 : 0;
        const int row = 4 * kk + c0;
        const int col = 16 * t + (ln & 15);
        v2f v;
        v.x = W1[(size_t)row       * Hh + col];
        v.y = W1[(size_t)(row + 1) * Hh + col];
        w1f[e] = v;                                            // ds_store_b64, lane-contiguous
    }
    for (int e = tid; e < NG * Dd; e += 256) sk[e]  = k_idx[gbase * Dd + e];
    for (int e = tid; e < Hh;      e += 256) sb1[e] = b1[e];
    for (int e = tid; e < Hh * Oo; e += 256) sw2[e] = W2[e];
    __syncthreads();

    const int wave   = tid >> 5;
    const int lane   = tid & 31;
    const int lane16 = lane & 15;
    const int khalf  = (lane < 16) ? 0 : 2;
    const int row0   = blockIdx.x * ROWS + wave * 16;

    const float* xrow = x + (size_t)(row0 + lane16) * (Gg * Dd);

    for (int gi = 0; gi < NG; ++gi) {
        const int  g   = gbase + gi;
        const int* skg = sk + gi * Dd;

        // ---- A fragments: gathered 16x128 fp32 x-tile, resident in 64 VGPRs ----
        // A layout (16x4, MxK): lanes 0-15 hold K=0,1; lanes 16-31 hold K=2,3; M = lane&15.
        v2f afrag[NK];
#pragma unroll
        for (int kk = 0; kk < NK; ++kk) {
            const int d = kk * 4 + khalf;
            afrag[kk].x = xrow[skg[d]];
            afrag[kk].y = xrow[skg[d + 1]];
        }

        float oacc[8][3];
#pragma unroll
        for (int r = 0; r < 8; ++r)
#pragma unroll
            for (int j = 0; j < 3; ++j) oacc[r][j] = 0.0f;

        // ---- 16 N-tiles: 32-deep f32 WMMA chain, B operands via conflict-free LDS b64 ----
        for (int t = 0; t < NT; ++t) {
            v8f c = {};
#pragma unroll
            for (int kk = 0; kk < NK; ++kk) {
                const v2f bfrag = w1f[(t * NK + kk) * 32 + lane];  // ds_load_b64
                c = __builtin_amdgcn_wmma_f32_16x16x4_f32(
                        /*neg_a=*/false, afrag[kk],
                        /*neg_b=*/false, bfrag,
                        /*c_mod=*/(short)0, c,
                        /*reuse_a=*/false, /*reuse_b=*/false);
            }

            // Fused bias + relu + tiny second GEMM (K=256, N=3) in VALU.
            // C/D layout: lanes 0-15 VGPR r -> M=r; lanes 16-31 VGPR r -> M=8+r; N = lane&15.
            const int   n   = t * 16 + lane16;
            const float b1v = sb1[n];
            const float w0  = sw2[n * Oo + 0];
            const float w1v = sw2[n * Oo + 1];
            const float w2v = sw2[n * Oo + 2];
#pragma unroll
            for (int r = 0; r < 8; ++r) {
                float h = c[r] + b1v;
                h = fmaxf(h, 0.0f);
                oacc[r][0] = fmaf(h, w0,  oacc[r][0]);
                oacc[r][1] = fmaf(h, w1v, oacc[r][1]);
                oacc[r][2] = fmaf(h, w2v, oacc[r][2]);
            }
        }

        // ---- Reduce partials across each 16-lane half-wave ----
#pragma unroll
        for (int r = 0; r < 8; ++r)
#pragma unroll
            for (int j = 0; j < 3; ++j) {
                float v = oacc[r][j];
                v += __shfl_xor(v, 1, 32);
                v += __shfl_xor(v, 2, 32);
                v += __shfl_xor(v, 4, 32);
                v += __shfl_xor(v, 8, 32);
                oacc[r][j] = v;
            }

        // ---- Scatter via v_idx: lanes 0..7 -> rows +0..7, lanes 16..23 -> rows +8..15 ----
        int r = -1, orow = 0;
        if (lane < 8)                     { r = lane;      orow = row0 + lane; }
        else if (lane >= 16 && lane < 24) { r = lane - 16; orow = row0 + 8 + (lane - 16); }
        if (r >= 0) {
#pragma unroll
            for (int j = 0; j < 3; ++j) {
                out[(size_t)orow * OUTC + v_idx[g * Oo + j]] = oacc[r][j] + b2[j];
            }
        }
    }
}

extern "C" void kernel_launch(void* const* d_in, const int* in_sizes, int n_in,
                              void* d_out, int out_size, void* d_ws, size_t ws_size,
                              hipStream_t stream) {
    const float* x     = (const float*)d_in[0];
    const float* W1    = (const float*)d_in[1];
    const float* b1    = (const float*)d_in[2];
    const float* W2    = (const float*)d_in[3];
    const float* b2    = (const float*)d_in[4];
    const int*   k_idx = (const int*)d_in[5];
    const int*   v_idx = (const int*)d_in[6];
    float*       out   = (float*)d_out;

    // Zero the output first: columns not covered by v_idx must stay 0.
    {
        int n = out_size;
        int blocks = (n + 255) / 256;
        zero_out_kernel<<<blocks, 256, 0, stream>>>(out, n);
    }

    dim3 grid(Bb / ROWS, Gg / NG);   // 128 x 5 = 640 blocks
    dim3 block(256);
    mlp_wmma_f32_kernel<<<grid, block, (int)LDS_TOT, stream>>>(
        x, W1, b1, W2, b2, k_idx, v_idx, out);
}